// WeightedAtomLayer_71313636983326
// MI455X (gfx1250) — compile-verified
//
#include <hip/hip_runtime.h>
#include <hip/hip_bf16.h>

typedef __attribute__((ext_vector_type(2))) float v2f;
typedef __attribute__((ext_vector_type(8))) float v8f;

#define NT      256          // number of atom types
#define D       64           // d_in == d_out == 64
#define TILE_M  16           // atoms per WMMA tile
#define GEMM_THREADS 128     // 4 waves of 32
#define TILE_GROUPS  8       // grid.y for the gemm kernel
#define LDSP    65           // padded LDS row stride (floats) to avoid bank conflicts

__device__ __forceinline__ float wal_tanh(float v) {
#if __has_builtin(__builtin_amdgcn_tanhf)
    return __builtin_amdgcn_tanhf(v);   // hardware V_TANH_F32 on gfx1250
#else
    return tanhf(v);
#endif
}

// ---------------- binning kernels ----------------

__global__ void wal_zero_counts(int* __restrict__ counts) {
    counts[threadIdx.x] = 0;
}

__global__ void wal_count(const int* __restrict__ types, int* __restrict__ counts, int n) {
    int i = blockIdx.x * blockDim.x + threadIdx.x;
    if (i < n) atomicAdd(&counts[types[i] & (NT - 1)], 1);
}

__global__ void wal_scan(const int* __restrict__ counts,
                         int* __restrict__ offsets,
                         int* __restrict__ cursors) {
    __shared__ int s_off[NT + 1];
    if (threadIdx.x == 0) {
        int acc = 0;
        for (int i = 0; i < NT; ++i) { s_off[i] = acc; acc += counts[i]; }
        s_off[NT] = acc;
    }
    __syncthreads();
    offsets[threadIdx.x] = s_off[threadIdx.x];
    cursors[threadIdx.x] = s_off[threadIdx.x];
    if (threadIdx.x == 0) offsets[NT] = s_off[NT];
}

__global__ void wal_scatter(const int* __restrict__ types,
                            int* __restrict__ cursors,
                            int* __restrict__ order, int n) {
    int i = blockIdx.x * blockDim.x + threadIdx.x;
    if (i < n) {
        int t = types[i] & (NT - 1);
        int pos = atomicAdd(&cursors[t], 1);
        order[pos] = i;
    }
}

// ---------------- WMMA GEMM kernel ----------------
// For type t: Y[16xD] = Xg[16xD] * W_t^T + b_t, then tanh.
// WMMA fragment layouts (CDNA5, wave32, V_WMMA_F32_16X16X4_F32):
//   A (16x4, MxK):  lane l holds A[l&15][2*(l>>4) + r], r = VGPR 0/1
//   B (4x16, KxN):  lane l holds B[2*(l>>4) + r][l&15]
//   C/D (16x16):    VGPR r, lane l -> D[r + 8*(l>>4)][l&15]
// B fragments are loop-invariant per block (one W_t per type), so they are
// preloaded into 32 VGPRs/lane once; the tile loop only streams A from LDS.
__global__ __launch_bounds__(GEMM_THREADS)
void wal_typed_linear_wmma(const float* __restrict__ x,
                           const float* __restrict__ W,
                           const float* __restrict__ b,
                           const int*   __restrict__ offsets,
                           const int*   __restrict__ order,
                           float*       __restrict__ out) {
    __shared__ float s_x[TILE_M * LDSP];   // gathered x tile, padded

    const int type = blockIdx.x;
    const int tid  = threadIdx.x;
    const int lane = tid & 31;
    const int wave = tid >> 5;
    const int half = lane >> 4;            // 0: lanes 0-15, 1: lanes 16-31
    const int lm   = lane & 15;
    const int nbase = wave * 16;           // 16-wide output slab per wave

    const int beg = offsets[type];
    const int cnt = offsets[type + 1] - beg;
    const int ntiles = (cnt + TILE_M - 1) / TILE_M;
    if (ntiles == 0) return;

    // Preload this wave's 16 B-fragments of W_t into registers (loop-invariant).
    // B[k][n] = W_t[n][k]  (y = x . W^T); per lane: row nbase+lm, cols kb*4+2*half.
    const float* __restrict__ wrow = W + ((size_t)type * D + (nbase + lm)) * D + 2 * half;
    v2f bfrag[16];
    #pragma unroll
    for (int kb = 0; kb < 16; ++kb) {
        bfrag[kb].x = wrow[kb * 4 + 0];
        bfrag[kb].y = wrow[kb * 4 + 1];
    }
    const float bias = b[type * D + nbase + lm];   // per-lane output-column bias

    const float4* __restrict__ x4 = (const float4*)x;   // row = 16 float4

    for (int t = blockIdx.y; t < ntiles; t += gridDim.y) {
        const int base  = beg + t * TILE_M;
        const int valid = cnt - t * TILE_M < TILE_M ? cnt - t * TILE_M : TILE_M;

        __syncthreads();                   // previous iteration done with s_x
        // Gather 16 atom rows as float4 (zero-fill partial tile so WMMA runs
        // with EXEC all-1s). order[] rows are L2-hot; read directly per thread.
        #pragma unroll
        for (int i = tid; i < TILE_M * 16; i += GEMM_THREADS) {
            const int r = i >> 4, c4 = i & 15;
            float4 v = make_float4(0.f, 0.f, 0.f, 0.f);
            if (r < valid) {
                v = x4[(size_t)order[base + r] * 16 + c4];
            }
            float* dst = &s_x[r * LDSP + c4 * 4];
            dst[0] = v.x; dst[1] = v.y; dst[2] = v.z; dst[3] = v.w;
        }
        __syncthreads();

        v8f acc = {};
        #pragma unroll
        for (int kb = 0; kb < 16; ++kb) {
            const int ko = kb * 4 + 2 * half;
            v2f a;
            a.x = s_x[lm * LDSP + ko];
            a.y = s_x[lm * LDSP + ko + 1];
            acc = __builtin_amdgcn_wmma_f32_16x16x4_f32(
                false, a, false, bfrag[kb], (short)0, acc, false, false);
        }

        const int n = nbase + lm;
        #pragma unroll
        for (int r = 0; r < 8; ++r) {
            const int m = r + 8 * half;
            if (m < valid) {
                out[(size_t)order[base + m] * D + n] = wal_tanh(acc[r] + bias);
            }
        }
    }
}

// ---------------- launch ----------------

extern "C" void kernel_launch(void* const* d_in, const int* in_sizes, int n_in,
                              void* d_out, int out_size, void* d_ws, size_t ws_size,
                              hipStream_t stream) {
    const float* x     = (const float*)d_in[0];
    const int*   types = (const int*)  d_in[1];   // harness delivers integers as int32
    const float* W     = (const float*)d_in[2];
    const float* b     = (const float*)d_in[3];
    float*       out   = (float*)d_out;
    const int n = in_sizes[1];                    // N_ATOMS

    char* ws      = (char*)d_ws;
    int*  counts  = (int*)(ws);            // 256 ints
    int*  cursors = (int*)(ws + 1024);     // 256 ints
    int*  offsets = (int*)(ws + 2048);     // 257 ints
    int*  order   = (int*)(ws + 4096);     // n ints

    wal_zero_counts<<<1, NT, 0, stream>>>(counts);
    wal_count<<<(n + 255) / 256, 256, 0, stream>>>(types, counts, n);
    wal_scan<<<1, NT, 0, stream>>>(counts, offsets, cursors);
    wal_scatter<<<(n + 255) / 256, 256, 0, stream>>>(types, cursors, order, n);

    dim3 grid(NT, TILE_GROUPS);
    wal_typed_linear_wmma<<<grid, GEMM_THREADS, 0, stream>>>(x, W, b, offsets, order, out);
}